// TimeSeriesGCN_63419487093297
// MI455X (gfx1250) — compile-verified
//
#include <hip/hip_runtime.h>
#include <hip/hip_bf16.h>
#include <cmath>

typedef __attribute__((ext_vector_type(2))) float v2f;
typedef __attribute__((ext_vector_type(8))) float v8f;

#define WMMA4(a, b, c) \
    __builtin_amdgcn_wmma_f32_16x16x4_f32(false, (a), false, (b), (short)0, (c), false, false)

// ---------------------------------------------------------------------------
// Degree / normalization
// ---------------------------------------------------------------------------
__global__ void k_init(float* __restrict__ deg, float* __restrict__ pooled, int N) {
    int i = blockIdx.x * 256 + threadIdx.x;
    if (i < N) deg[i] = 1.0f;                       // self-loop
    if (i < 64) pooled[i] = -3.402823466e38f;       // -inf for max pool
}

__global__ void k_deg(const long long* __restrict__ dst, float* __restrict__ deg, int E) {
    int e = blockIdx.x * 256 + threadIdx.x;
    if (e < E) atomicAdd(&deg[(int)dst[e]], 1.0f);
}

__global__ void k_rsqrt(float* __restrict__ deg, int N) {
    int i = blockIdx.x * 256 + threadIdx.x;
    if (i < N) deg[i] = rsqrtf(deg[i]);             // in place: deg -> dinv
}

// ---------------------------------------------------------------------------
// GEMM1: h1 = x[ N x 256 ] @ W1[ 256 x 16 ]   (f32 WMMA 16x16x4)
// block = 256 threads = 8 waves, each wave owns a 16-row tile (128 rows/block).
// A-loads use clamped row indices (branch-free); stores take a tile-uniform
// fast path when the whole 16-row tile is in range (all blocks but the last).
// ---------------------------------------------------------------------------
__global__ void k_gemm1(const float* __restrict__ x, const float* __restrict__ W1,
                        float* __restrict__ h1, int N) {
    __shared__ float sW[256 * 16];                  // 16 KB
    const int t = threadIdx.x;
    for (int i = t; i < 256 * 16; i += 256) sW[i] = W1[i];
    __syncthreads();

    const int lane  = t & 31;
    const int wave  = t >> 5;
    const int base  = blockIdx.x * 128 + wave * 16;
    const int mrow  = lane & 15;
    const int khalf = (lane >> 4) << 1;             // 0 or 2
    const int node  = base + mrow;
    const int nclmp = node < N ? node : (N - 1);    // branch-free OOB handling
    const float* xrow = x + (size_t)nclmp * 256;

    // warm the row's cachelines (global_prefetch_b8)
    __builtin_prefetch(xrow, 0, 0);
    __builtin_prefetch(xrow + 128, 0, 0);

    v8f c = {};
    #pragma unroll 8
    for (int kk = 0; kk < 64; ++kk) {
        const int k0 = kk * 4 + khalf;
        v2f a, b;
        a.x = xrow[k0];
        a.y = xrow[k0 + 1];
        b.x = sW[k0 * 16 + mrow];
        b.y = sW[(k0 + 1) * 16 + mrow];
        c = WMMA4(a, b, c);
    }
    const int rbase = (lane >> 4) * 8;              // C: vgpr r -> row r + 8*(lane>=16)
    float* hrow = h1 + (size_t)(base + rbase) * 16 + mrow;
    if (base + 16 <= N) {                           // tile fully in range: no guards
        #pragma unroll
        for (int r = 0; r < 8; ++r) hrow[r * 16] = c[r];
    } else {
        #pragma unroll
        for (int r = 0; r < 8; ++r)
            if (base + rbase + r < N) hrow[r * 16] = c[r];
    }
}

// ---------------------------------------------------------------------------
// Self-loop term (also zero-initializes agg) + edge scatter
// ---------------------------------------------------------------------------
__global__ void k_self(const float* __restrict__ h, const float* __restrict__ dinv,
                       float* __restrict__ agg, int total, int logF) {
    int idx = blockIdx.x * 256 + threadIdx.x;
    if (idx < total) {
        int n = idx >> logF;
        float d = dinv[n];
        agg[idx] = h[idx] * d * d;
    }
}

__global__ void k_scat(const float* __restrict__ h, const long long* __restrict__ src,
                       const long long* __restrict__ dst, const float* __restrict__ dinv,
                       float* __restrict__ agg, int E, int logF) {
    int idx = blockIdx.x * 256 + threadIdx.x;
    int total = E << logF;
    if (idx < total) {
        int e = idx >> logF;
        int f = idx & ((1 << logF) - 1);
        int s = (int)src[e];
        int d = (int)dst[e];
        float w = dinv[s] * dinv[d];
        atomicAdd(agg + (((size_t)d) << logF) + f, h[(((size_t)s) << logF) + f] * w);
    }
}

// ---------------------------------------------------------------------------
// GEMM2: h2 = relu(agg1 + b1)[ N x 16 ] @ W2[ 16 x 32 ]  (bias+ReLU fused in A)
// ---------------------------------------------------------------------------
__global__ void k_gemm2(const float* __restrict__ agg1, const float* __restrict__ W2,
                        const float* __restrict__ b1, float* __restrict__ h2, int N) {
    __shared__ float sW[16 * 32];                   // 2 KB
    __shared__ float sb1[16];
    const int t = threadIdx.x;
    for (int i = t; i < 16 * 32; i += 256) sW[i] = W2[i];
    if (t < 16) sb1[t] = b1[t];
    __syncthreads();

    const int lane  = t & 31;
    const int wave  = t >> 5;
    const int base  = blockIdx.x * 128 + wave * 16;
    const int mrow  = lane & 15;
    const int khalf = (lane >> 4) << 1;
    const int node  = base + mrow;
    const int nclmp = node < N ? node : (N - 1);
    const float* arow = agg1 + (size_t)nclmp * 16;

    v8f c0 = {}, c1 = {};
    #pragma unroll
    for (int kk = 0; kk < 4; ++kk) {
        const int k0 = kk * 4 + khalf;
        v2f a, bA, bB;
        a.x = fmaxf(arow[k0] + sb1[k0], 0.f);
        a.y = fmaxf(arow[k0 + 1] + sb1[k0 + 1], 0.f);
        bA.x = sW[k0 * 32 + mrow];        bA.y = sW[(k0 + 1) * 32 + mrow];
        bB.x = sW[k0 * 32 + 16 + mrow];   bB.y = sW[(k0 + 1) * 32 + 16 + mrow];
        c0 = WMMA4(a, bA, c0);
        c1 = WMMA4(a, bB, c1);
    }
    const int rbase = (lane >> 4) * 8;
    float* hrow = h2 + (size_t)(base + rbase) * 32 + mrow;
    if (base + 16 <= N) {
        #pragma unroll
        for (int r = 0; r < 8; ++r) {
            hrow[r * 32]      = c0[r];
            hrow[r * 32 + 16] = c1[r];
        }
    } else {
        #pragma unroll
        for (int r = 0; r < 8; ++r) {
            if (base + rbase + r < N) {
                hrow[r * 32]      = c0[r];
                hrow[r * 32 + 16] = c1[r];
            }
        }
    }
}

// ---------------------------------------------------------------------------
// Conv2d(32->64, kernel (1,3), pad 1) as K=96 WMMA GEMM, fused global max pool.
// Zero-padding at the graph boundary via clamped address + mask multiply.
// ---------------------------------------------------------------------------
__device__ inline void atomicMaxF(float* addr, float val) {
    int* ai = (int*)addr;
    int old = *ai;
    while (__int_as_float(old) < val) {
        int assumed = old;
        old = atomicCAS(ai, assumed, __float_as_int(val));
        if (old == assumed) break;
    }
}

__device__ __forceinline__ void tile_max_reduce(v8f cv, float cb, int base, int rbase,
                                                int N, bool full, float* __restrict__ pooled,
                                                int col, int lane) {
    float m = -3.402823466e38f;
    if (full) {
        #pragma unroll
        for (int r = 0; r < 8; ++r) m = fmaxf(m, cv[r] + cb);
    } else {
        #pragma unroll
        for (int r = 0; r < 8; ++r)
            if (base + rbase + r < N) m = fmaxf(m, cv[r] + cb);
    }
    m = fmaxf(m, __shfl_xor(m, 16, 32));            // combine the two half-waves
    if (lane < 16) atomicMaxF(&pooled[col], m);
}

__global__ void k_conv_pool(const float* __restrict__ agg2, const float* __restrict__ b2,
                            const float* __restrict__ conv_w, const float* __restrict__ conv_b,
                            float* __restrict__ pooled, int N) {
    __shared__ float sW[96 * 64];                   // 24 KB, sW[k][o], k = tap*32 + i
    __shared__ float sb2[32];
    const int t = threadIdx.x;
    for (int idx = t; idx < 96 * 64; idx += 256) {
        int k = idx >> 6, o = idx & 63;
        int tap = k >> 5, i = k & 31;
        sW[idx] = conv_w[o * 96 + i * 3 + tap];     // OIHW [64,32,1,3]
    }
    if (t < 32) sb2[t] = b2[t];
    __syncthreads();

    const int lane  = t & 31;
    const int wave  = t >> 5;
    const int base  = blockIdx.x * 128 + wave * 16;
    const int mrow  = lane & 15;
    const int khalf = (lane >> 4) << 1;

    v8f c0 = {}, c1 = {}, c2 = {}, c3 = {};
    #pragma unroll 4
    for (int kk = 0; kk < 24; ++kk) {
        const int k0 = kk * 4 + khalf;
        v2f a;
        #pragma unroll
        for (int j = 0; j < 2; ++j) {
            const int k  = k0 + j;
            const int nn = base + mrow + (k >> 5) - 1;       // tap-1 node offset
            const int nc = nn < 0 ? 0 : (nn >= N ? N - 1 : nn);
            const float msk = (nn >= 0 && nn < N) ? 1.f : 0.f;
            const float v = (agg2[(size_t)nc * 32 + (k & 31)] + sb2[k & 31]) * msk;
            if (j == 0) a.x = v; else a.y = v;
        }
        v2f b0, b1v, b2v, b3;
        b0.x  = sW[k0 * 64 + mrow];        b0.y  = sW[(k0 + 1) * 64 + mrow];
        b1v.x = sW[k0 * 64 + 16 + mrow];   b1v.y = sW[(k0 + 1) * 64 + 16 + mrow];
        b2v.x = sW[k0 * 64 + 32 + mrow];   b2v.y = sW[(k0 + 1) * 64 + 32 + mrow];
        b3.x  = sW[k0 * 64 + 48 + mrow];   b3.y  = sW[(k0 + 1) * 64 + 48 + mrow];
        c0 = WMMA4(a, b0, c0);
        c1 = WMMA4(a, b1v, c1);
        c2 = WMMA4(a, b2v, c2);
        c3 = WMMA4(a, b3, c3);
    }

    const int  rbase = (lane >> 4) * 8;
    const bool full  = (base + 16 <= N);
    tile_max_reduce(c0, conv_b[mrow],      base, rbase, N, full, pooled, mrow,      lane);
    tile_max_reduce(c1, conv_b[16 + mrow], base, rbase, N, full, pooled, 16 + mrow, lane);
    tile_max_reduce(c2, conv_b[32 + mrow], base, rbase, N, full, pooled, 32 + mrow, lane);
    tile_max_reduce(c3, conv_b[48 + mrow], base, rbase, N, full, pooled, 48 + mrow, lane);
}

// ---------------------------------------------------------------------------
// FC: out[c] = fc_b[c] + sum_o pooled[o] * fc_w[c*64+o]
// ---------------------------------------------------------------------------
__global__ void k_fc(const float* __restrict__ pooled, const float* __restrict__ fc_w,
                     const float* __restrict__ fc_b, float* __restrict__ out, int C) {
    int c = blockIdx.x * 256 + threadIdx.x;
    if (c >= C) return;
    float s = fc_b[c];
    #pragma unroll 8
    for (int o = 0; o < 64; ++o) s += pooled[o] * fc_w[c * 64 + o];
    out[c] = s;
}

// ---------------------------------------------------------------------------
extern "C" void kernel_launch(void* const* d_in, const int* in_sizes, int n_in,
                              void* d_out, int out_size, void* d_ws, size_t ws_size,
                              hipStream_t stream) {
    const float*     x      = (const float*)d_in[0];
    const long long* edge   = (const long long*)d_in[1];   // int64 [2, E]
    const float*     W1     = (const float*)d_in[2];
    const float*     b1     = (const float*)d_in[3];
    const float*     W2     = (const float*)d_in[4];
    const float*     b2     = (const float*)d_in[5];
    const float*     conv_w = (const float*)d_in[6];
    const float*     conv_b = (const float*)d_in[7];
    const float*     fc_w   = (const float*)d_in[8];
    const float*     fc_b   = (const float*)d_in[9];
    float*           out    = (float*)d_out;

    const int N = in_sizes[0] / 256;
    const int E = in_sizes[1] / 2;
    const long long* src = edge;
    const long long* dst = edge + E;

    float* ws     = (float*)d_ws;
    float* dinv   = ws;                              // N  (deg, then rsqrt in place)
    float* h1     = dinv + N;                        // 16N
    float* agg1   = h1   + (size_t)16 * N;           // 16N
    float* h2     = agg1 + (size_t)16 * N;           // 32N
    float* agg2   = h2   + (size_t)32 * N;           // 32N
    float* pooled = agg2 + (size_t)32 * N;           // 64

    const dim3 B(256);

    k_init  <<<dim3((N + 255) / 256),        B, 0, stream>>>(dinv, pooled, N);
    k_deg   <<<dim3((E + 255) / 256),        B, 0, stream>>>(dst, dinv, E);
    k_rsqrt <<<dim3((N + 255) / 256),        B, 0, stream>>>(dinv, N);

    k_gemm1 <<<dim3((N + 127) / 128),        B, 0, stream>>>(x, W1, h1, N);
    k_self  <<<dim3((N * 16 + 255) / 256),   B, 0, stream>>>(h1, dinv, agg1, N * 16, 4);
    k_scat  <<<dim3(((E << 4) + 255) / 256), B, 0, stream>>>(h1, src, dst, dinv, agg1, E, 4);

    k_gemm2 <<<dim3((N + 127) / 128),        B, 0, stream>>>(agg1, W2, b1, h2, N);
    k_self  <<<dim3((N * 32 + 255) / 256),   B, 0, stream>>>(h2, dinv, agg2, N * 32, 5);
    k_scat  <<<dim3(((E << 5) + 255) / 256), B, 0, stream>>>(h2, src, dst, dinv, agg2, E, 5);

    k_conv_pool <<<dim3((N + 127) / 128),    B, 0, stream>>>(agg2, b2, conv_w, conv_b, pooled, N);
    k_fc        <<<dim3((out_size + 255) / 256), B, 0, stream>>>(pooled, fc_w, fc_b, out, out_size);
}